// ForwardTransmission_89541478187334
// MI455X (gfx1250) — compile-verified
//
#include <hip/hip_runtime.h>
#include <hip/hip_bf16.h>
#include <stdint.h>
#include <stddef.h>

// ---------------- problem constants (match reference) ----------------
constexpr int   BB = 2048;   // batch
constexpr int   TT = 2048;   // timesteps (serial scan)
constexpr int   HH = 7;      // hidden units per MLP
constexpr int   CC = 9;      // number of MLP components
constexpr float FS_F = 3051.76f;

// ---------------- tiling ----------------
constexpr int ROWS   = 32;        // batch rows per block == wave32
constexpr int TC     = 32;        // timesteps per LDS chunk
constexpr int STRIDE = TC + 1;    // 33 dwords: gcd(33,64)=1 -> conflict-free LDS
constexpr int NCH    = TT / TC;   // 64 chunks
constexpr int NW     = CC * HH;   // 63 weights per tensor

// ---------------- gfx1250 feature probes ----------------
#if __has_builtin(__builtin_amdgcn_global_load_async_to_lds_b32) && \
    __has_builtin(__builtin_amdgcn_s_wait_asynccnt)
#define HAVE_ASYNC 1
#else
#define HAVE_ASYNC 0
#endif

typedef __attribute__((address_space(1))) int as1_i32;
typedef __attribute__((address_space(3))) int as3_i32;

__device__ __forceinline__ float tanh_hw(float x) {
#if __has_builtin(__builtin_amdgcn_tanhf)
  return __builtin_amdgcn_tanhf(x);                 // v_tanh_f32 (CDNA5 TRANS op)
#elif __has_builtin(__builtin_amdgcn_tanh_f32)
  return __builtin_amdgcn_tanh_f32(x);
#else
  float r;
  asm volatile("v_tanh_f32 %0, %1\n\tv_nop" : "=v"(r) : "v"(x));
  return r;
#endif
}

__device__ __forceinline__ void async_ld_f32(const float* g, float* l) {
#if HAVE_ASYNC
  __builtin_amdgcn_global_load_async_to_lds_b32(
      (as1_i32*)g, (as3_i32*)l, /*imm offset*/0, /*cpol*/0);
#else
  *l = *g;
#endif
}

template <int N>
__device__ __forceinline__ void wait_async() {
#if HAVE_ASYNC
  __builtin_amdgcn_s_wait_asynccnt(N);              // s_wait_asynccnt N
#endif
}

// one tiny MLP, H=7, all operands VGPR-resident; two accumulation chains
// to shorten the serial dependency path (TRANS/VALU co-execution friendly).
__device__ __forceinline__ float mlp7(float v, const float* wi, const float* bi,
                                      const float* wo, float bo) {
  float t0 = tanh_hw(__builtin_fmaf(v, wi[0], bi[0]));
  float t1 = tanh_hw(__builtin_fmaf(v, wi[1], bi[1]));
  float t2 = tanh_hw(__builtin_fmaf(v, wi[2], bi[2]));
  float t3 = tanh_hw(__builtin_fmaf(v, wi[3], bi[3]));
  float t4 = tanh_hw(__builtin_fmaf(v, wi[4], bi[4]));
  float t5 = tanh_hw(__builtin_fmaf(v, wi[5], bi[5]));
  float t6 = tanh_hw(__builtin_fmaf(v, wi[6], bi[6]));
  float a0 = __builtin_fmaf(wo[0], t0, bo);
  float a1 = wo[1] * t1;
  a0 = __builtin_fmaf(wo[2], t2, a0);
  a1 = __builtin_fmaf(wo[3], t3, a1);
  a0 = __builtin_fmaf(wo[4], t4, a0);
  a1 = __builtin_fmaf(wo[5], t5, a1);
  a0 = __builtin_fmaf(wo[6], t6, a0);
  return a0 + a1;
}

__global__ __launch_bounds__(ROWS)
void ForwardTransmission_89541478187334_kernel(const float* __restrict__ x,
                                               const float* __restrict__ w_in,
                                               const float* __restrict__ b_in,
                                               const float* __restrict__ w_out,
                                               const float* __restrict__ b_out,
                                               float* __restrict__ out) {
  __shared__ float xt_s[2][ROWS * STRIDE];  // double-buffered input tiles
  __shared__ float ot_s[ROWS * STRIDE];     // output staging tile
  __shared__ float w_s[3 * NW + CC];        // weight bounce buffer (forces VGPRs)

  const int lane = threadIdx.x;             // 0..31 (wave32)
  const int b0   = blockIdx.x * ROWS;       // batch base for this block/wave

  // ---- prologue: async-stream chunk 0 into LDS (coalesced 128B rows) ----
#pragma unroll
  for (int r = 0; r < ROWS; ++r)
    async_ld_f32(x + (size_t)(b0 + r) * TT + lane, &xt_s[0][r * STRIDE + lane]);

  // ---- stage weights through LDS so they become VGPR-resident.
  // Loading them directly would scalarize into SGPRs (observed: SGPR spills +
  // per-iteration v_readlane restores). ds_load results are always VGPRs.
#pragma unroll
  for (int i = lane; i < NW; i += ROWS) {   // coalesced global reads, 2 per lane
    w_s[i]          = w_in [i];
    w_s[NW + i]     = b_in [i];
    w_s[2 * NW + i] = w_out[i];
  }
  if (lane < CC) w_s[3 * NW + lane] = b_out[lane];
  // single wave: ds_store -> ds_load ordering is handled by DScnt in-order.

  float wi[CC][HH], bi[CC][HH], wo[CC][HH], bo[CC];
#pragma unroll
  for (int c = 0; c < CC; ++c) {
#pragma unroll
    for (int h = 0; h < HH; ++h) {
      wi[c][h] = w_s[c * HH + h];           // uniform-address LDS broadcast
      bi[c][h] = w_s[NW + c * HH + h];
      wo[c][h] = w_s[2 * NW + c * HH + h];
    }
    bo[c] = w_s[3 * NW + c];
  }

  float q1 = 0.f, q2 = 0.f, q3 = 0.f;
  const float invfs = 1.0f / FS_F;

  for (int ch = 0; ch < NCH; ++ch) {
    // prefetch next chunk, then wait for the oldest 32 (current chunk) loads;
    // async loads complete in order, so asynccnt<=32 => chunk ch is in LDS.
    if (ch + 1 < NCH) {
      const int t0n = (ch + 1) * TC;
      float* buf = xt_s[(ch + 1) & 1];
#pragma unroll
      for (int r = 0; r < ROWS; ++r)
        async_ld_f32(x + (size_t)(b0 + r) * TT + t0n + lane,
                     &buf[r * STRIDE + lane]);
      wait_async<ROWS>();
    } else {
      wait_async<0>();
    }

    const float* xs = &xt_s[ch & 1][lane * STRIDE];  // stride-33: no conflicts
    float*       os = &ot_s[lane * STRIDE];

#pragma unroll 1
    for (int t = 0; t < TC; ++t) {
      const float xt = xs[t];

      const float V1 = tanh_hw(mlp7(q1,            wi[0], bi[0], wo[0], bo[0]));
      const float V2 = tanh_hw(mlp7(q2,            wi[1], bi[1], wo[1], bo[1]));
      const float V3 = tanh_hw(mlp7(q3,            wi[2], bi[2], wo[2], bo[2]));
      const float I1 = tanh_hw(mlp7(V1,            wi[3], bi[3], wo[3], bo[3]));
      const float I2 = tanh_hw(mlp7(V2,            wi[4], bi[4], wo[4], bo[4]));
      const float I3 = tanh_hw(mlp7(V3,            wi[5], bi[5], wo[5], bo[5]));
      const float I5 = tanh_hw(mlp7(V2 - V3,       wi[6], bi[6], wo[6], bo[6]));
      const float IT = tanh_hw(mlp7(xt - V1 - V2,  wi[7], bi[7], wo[7], bo[7]));

      q1 += (IT - I1)      * invfs;
      q2 += (IT - I2 - I5) * invfs;
      q3 += (I5 - I3)      * invfs;

      os[t] = mlp7(IT, wi[8], bi[8], wo[8], bo[8]);  // RICME: no final tanh
    }

    // flush staged outputs, coalesced along t (conflict-free LDS reads)
    const int t0 = ch * TC;
#pragma unroll
    for (int r = 0; r < ROWS; ++r)
      out[(size_t)(b0 + r) * TT + t0 + lane] = ot_s[r * STRIDE + lane];
  }
}

extern "C" void kernel_launch(void* const* d_in, const int* in_sizes, int n_in,
                              void* d_out, int out_size, void* d_ws, size_t ws_size,
                              hipStream_t stream) {
  (void)in_sizes; (void)n_in; (void)out_size; (void)d_ws; (void)ws_size;
  const float* x     = (const float*)d_in[0];
  const float* w_in  = (const float*)d_in[1];
  const float* b_in  = (const float*)d_in[2];
  const float* w_out = (const float*)d_in[3];
  const float* b_out = (const float*)d_in[4];
  float*       out   = (float*)d_out;

  dim3 grid(BB / ROWS);   // 64 blocks -> 64 WGPs busy
  dim3 block(ROWS);       // one wave32 per block, no barriers needed
  hipLaunchKernelGGL(ForwardTransmission_89541478187334_kernel,
                     grid, block, 0, stream, x, w_in, b_in, w_out, b_out, out);
}